// GraphEncoder_86543591015006
// MI455X (gfx1250) — compile-verified
//
#include <hip/hip_runtime.h>

// ---------------- problem constants (match reference) ----------------
#define EL   8000000
#define NL   1000000
#define NLP  1000064          // NL padded to multiple of 128 (zeroed pad rows)
#define EG   1000000
#define NG   65536
#define DD   128
#define NB_  506
#define SEQ_ 512

typedef float v2f __attribute__((ext_vector_type(2)));
typedef float v8f __attribute__((ext_vector_type(8)));

// f32 WMMA: D = A(16x4) * B(4x16) + C(16x16), wave32.
// A per lane: m = lane&15, K = 2*(lane>>4) + {0,1} across the 2 VGPRs.
// B per lane: n = lane&15, K = vgpr + 2*(lane>>4).
// C/D per lane: m = vgpr + 8*(lane>>4), n = lane&15.
__device__ __forceinline__ v8f wmma4(v2f a, v2f b, v8f c) {
  return __builtin_amdgcn_wmma_f32_16x16x4_f32(false, a, false, b, (short)0, c, false, false);
}

// ---------------- workspace layout (floats) ----------------
#define WS_AGG_L  0L
#define WS_AGG_G  12000768L            // NLP*12
#define WS_POOL   12787200L            // + NG*12
#define WS_CNT    12851968L            // + 506*128
#define WS_GSUM   12852480L            // + 512
#define WS_PADD   12852608L            // + 128
#define WS_FSWZ   12852736L            // + 128      (fused, pair-swizzled 512x128)
#define WS_SPW    12918272L            // + 512*128  (seqproj_w^T padded+swizzled 512x512)
#define WS_ZERO_N 12852608L            // zero accumulators: agg_l, agg_g, pool, cnt, gsum

__global__ void zero_ws_kernel(float* ws, long n) {
  long i = (long)blockIdx.x * blockDim.x + threadIdx.x;
  long stride = (long)gridDim.x * blockDim.x;
  for (; i < n; i += stride) ws[i] = 0.0f;
}

// ---------------- per-edge GINE message + scatter ----------------
// msg = relu(1 + [w,kmer] @ lin_w + lin_b); atomic add into agg[dst, 0..8] (stride 12)
__global__ void edge_msg_kernel(const int* __restrict__ eidx,
                                const float* __restrict__ wgt,
                                const float* __restrict__ kmer,
                                const float* __restrict__ linw,
                                const float* __restrict__ linb,
                                float* __restrict__ agg, int E) {
  __shared__ float lw[81];
  __shared__ float lb[9];
  int tid = threadIdx.x;
  if (tid < 81) lw[tid] = linw[tid];
  if (tid < 9)  lb[tid] = linb[tid];
  __syncthreads();
  int e = blockIdx.x * blockDim.x + tid;
  if (e >= E) return;
  float a[9];
  a[0] = wgt[e];
  float4 k0v = *(const float4*)(kmer + (long)e * 8);
  float4 k1v = *(const float4*)(kmer + (long)e * 8 + 4);
  a[1] = k0v.x; a[2] = k0v.y; a[3] = k0v.z; a[4] = k0v.w;
  a[5] = k1v.x; a[6] = k1v.y; a[7] = k1v.z; a[8] = k1v.w;
  int dst = eidx[E + e];                 // edge_index row 1
  float* dp = agg + (long)dst * 12;
  #pragma unroll
  for (int c = 0; c < 9; ++c) {
    float s = lb[c];
    #pragma unroll
    for (int r = 0; r < 9; ++r) s += a[r] * lw[r * 9 + c];
    s = 1.0f + s;
    s = s > 0.0f ? s : 0.0f;
    atomicAdd(dp + c, s);
  }
}

// ---------------- batch counts (LDS histogram) ----------------
__global__ void count_batch_kernel(const int* __restrict__ batch, float* __restrict__ cnt,
                                   long N, int nb) {
  __shared__ float h[512];
  int tid = threadIdx.x;
  for (int i = tid; i < 512; i += blockDim.x) h[i] = 0.0f;
  __syncthreads();
  for (long i = (long)blockIdx.x * blockDim.x + tid; i < N; i += (long)gridDim.x * blockDim.x)
    atomicAdd(&h[batch[i]], 1.0f);
  __syncthreads();
  for (int i = tid; i < nb; i += blockDim.x) atomicAdd(&cnt[i], h[i]);
}

// ---------------- node MLP: relu(h@W1+b1)@W2+b2, leaky, pool ----------------
// MODE 0: local (segment-mean by sorted batch); MODE 1: global (sum all -> gsum)
// Weights are staged in LDS pair-swizzled: w[kpair][n][2] so every B fragment
// (two K values for one column) is a single ds_load_b64.
template <int MODE>
__global__ void __launch_bounds__(256)
node_mlp_kernel(const float* __restrict__ agg,
                const float* __restrict__ w1, const float* __restrict__ b1,
                const float* __restrict__ w2, const float* __restrict__ b2,
                const int* __restrict__ batch,
                float* __restrict__ pool, float* __restrict__ gsum,
                int N, int nb) {
  __shared__ float w1s[6 * 256];        // 12(K,pad from 9) x 128, swizzled
  __shared__ float b1s[128];
  __shared__ float b2s[128];
  __shared__ float w2s[64 * 256];       // 128 x 128, swizzled (64 KB)
  __shared__ float hid[8 * 16 * 128];   // per-wave 16x128 hidden (64 KB)
  __shared__ float pacc[8 * 128];       // local: 8 batch slots; global: gacc in [0..127]
  __shared__ int   sbfirst;

  int tid = threadIdx.x;
  // stage W1 swizzled (pad rows 9..11 = 0)
  for (int i = tid; i < 12 * 128; i += 256) {
    int k = i >> 7, n = i & 127;
    w1s[(k >> 1) * 256 + n * 2 + (k & 1)] = (k < 9) ? w1[k * 128 + n] : 0.0f;
  }
  // stage W2 swizzled: one b64 LDS store per K-pair
  for (int i = tid; i < 64 * 128; i += 256) {
    int kp = i >> 7, n = i & 127;
    v2f val;
    val.x = w2[(2 * kp)     * 128 + n];
    val.y = w2[(2 * kp + 1) * 128 + n];
    *(v2f*)(w2s + kp * 256 + n * 2) = val;
  }
  if (tid < 128) { b1s[tid] = b1[tid]; b2s[tid] = b2[tid]; }
  for (int i = tid; i < 8 * 128; i += 256) pacc[i] = 0.0f;
  int blockBase = blockIdx.x * 128;
  if (MODE == 0 && tid == 0) sbfirst = batch[blockBase];
  __syncthreads();

  int wave = tid >> 5, lane = tid & 31;
  int m = lane & 15, g = lane >> 4;     // m: A row / B,C col index; g: lane-half
  int n2 = m * 2;                       // swizzled column offset
  int nodeBase = blockBase + wave * 16;
  int nodeA = nodeBase + m;             // may run into zeroed pad rows; masked at store

  // ---- layer 1: [16 x 12] @ [12 x 128], 3 k-steps x 8 n-tiles
  // A cols 9..11 pair with zero W1 rows, so "1 + agg" on pad cols is harmless.
  v8f acc[8] = {};
  const float* aggrow = agg + (long)nodeA * 12;
  #pragma unroll
  for (int kb = 0; kb < 12; kb += 4) {
    v2f a = *(const v2f*)(aggrow + kb + 2 * g);
    a.x += 1.0f; a.y += 1.0f;
    #pragma unroll
    for (int nt = 0; nt < 8; ++nt) {
      v2f bf = *(const v2f*)(w1s + ((kb >> 1) + g) * 256 + nt * 32 + n2);
      acc[nt] = wmma4(a, bf, acc[nt]);
    }
  }
  float* hw = hid + wave * 2048;
  #pragma unroll
  for (int nt = 0; nt < 8; ++nt) {
    #pragma unroll
    for (int v = 0; v < 8; ++v) {
      int mm = v + 8 * g;
      int n  = nt * 16 + m;
      float val = acc[nt][v] + b1s[n];
      hw[mm * 128 + n] = val > 0.0f ? val : 0.0f;
    }
  }
  __syncthreads();

  // ---- layer 2: [16 x 128] @ [128 x 128], software-pipelined b64 LDS loads
  v8f acc2[8] = {};
  const float* arow = hw + m * 128;
  v2f a_c = *(const v2f*)(arow + 2 * g);
  v2f b_c[8];
  #pragma unroll
  for (int nt = 0; nt < 8; ++nt)
    b_c[nt] = *(const v2f*)(w2s + g * 256 + nt * 32 + n2);
  for (int kb = 0; kb < 124; kb += 4) {
    int kp = (kb >> 1) + 2 + g;         // pair index of k-step kb+4
    v2f a_n = *(const v2f*)(arow + kb + 4 + 2 * g);
    v2f b_n[8];
    #pragma unroll
    for (int nt = 0; nt < 8; ++nt)
      b_n[nt] = *(const v2f*)(w2s + kp * 256 + nt * 32 + n2);
    #pragma unroll
    for (int nt = 0; nt < 8; ++nt) acc2[nt] = wmma4(a_c, b_c[nt], acc2[nt]);
    a_c = a_n;
    #pragma unroll
    for (int nt = 0; nt < 8; ++nt) b_c[nt] = b_n[nt];
  }
  #pragma unroll
  for (int nt = 0; nt < 8; ++nt) acc2[nt] = wmma4(a_c, b_c[nt], acc2[nt]);

  // ---- epilogue: bias + leaky_relu + pooling (batch load hoisted per row)
  #pragma unroll
  for (int v = 0; v < 8; ++v) {
    int mm = v + 8 * g;
    int nd = nodeBase + mm;
    bool ok = nd < N;
    int bo = 0; long prow = 0;
    if (MODE == 0 && ok) {
      int bb = batch[nd];
      bo = bb - sbfirst;                // batch sorted -> usually < 8
      prow = (long)bb * 128;
    }
    #pragma unroll
    for (int nt = 0; nt < 8; ++nt) {
      int n = nt * 16 + m;
      float val = acc2[nt][v] + b2s[n];
      val = val > 0.0f ? val : 0.01f * val;
      if (ok) {
        if (MODE == 0) {
          if ((unsigned)bo < 8u) atomicAdd(&pacc[bo * 128 + n], val);
          else                   atomicAdd(&pool[prow + n], val);
        } else {
          atomicAdd(&pacc[n], val);
        }
      }
    }
  }
  __syncthreads();
  if (MODE == 0) {
    for (int i = tid; i < 8 * 128; i += 256) {
      int bb = sbfirst + (i >> 7);
      if (bb < nb) atomicAdd(&pool[(long)bb * 128 + (i & 127)], pacc[i]);
    }
  } else {
    if (tid < 128) atomicAdd(&gsum[tid], pacc[tid]);
  }
}

// ---------------- constant attention vector ----------------
// softmax over 1 key == 1 => attn = v = g@Wv+bv; padd = ((v@Wo+bo)@Pw + pb)
__global__ void fuse_vec_kernel(const float* __restrict__ gsum,
                                const float* __restrict__ wv, const float* __restrict__ bv,
                                const float* __restrict__ wo, const float* __restrict__ bo,
                                const float* __restrict__ pw, const float* __restrict__ pb,
                                float* __restrict__ padd) {
  __shared__ float ge[128], vv[128], ao[128];
  int d = threadIdx.x;                        // 128 threads
  ge[d] = gsum[d] * (1.0f / (float)NG);
  __syncthreads();
  float s = bv[d];
  for (int k = 0; k < 128; ++k) s += ge[k] * wv[k * 128 + d];
  vv[d] = s; __syncthreads();
  s = bo[d];
  for (int k = 0; k < 128; ++k) s += vv[k] * wo[k * 128 + d];
  ao[d] = s; __syncthreads();
  s = pb[d];
  for (int k = 0; k < 128; ++k) s += ao[k] * pw[k * 128 + d];
  padd[d] = s;
}

// fused[b,d] = pool[b,d]/max(cnt,1) + padd[d], written pair-swizzled for the GEMM;
// rows [506,512) zeroed (K padding); attn_weights == 1.0 (softmax over 1 key)
__global__ void fuse_rows_kernel(const float* __restrict__ pool, const float* __restrict__ cnt,
                                 const float* __restrict__ padd,
                                 float* __restrict__ fswz, float* __restrict__ attnw) {
  int i = blockIdx.x * blockDim.x + threadIdx.x;   // 0 .. 512*128-1
  int b = i >> 7, d = i & 127;
  float val = 0.0f;
  if (b < NB_) {
    float c = cnt[b];
    c = c > 1.0f ? c : 1.0f;
    val = pool[i] / c + padd[d];
    if (d == 0) attnw[b] = 1.0f;
  }
  fswz[(b >> 1) * 256 + d * 2 + (b & 1)] = val;
}

// seqproj_w^T, zero-padded K 506->512, pair-swizzled: swz[kpair][s][2]
__global__ void spw_swz_kernel(const float* __restrict__ spw, float* __restrict__ swz) {
  int i = blockIdx.x * blockDim.x + threadIdx.x;   // 0 .. 512*512-1
  int k = i >> 9, s = i & 511;
  swz[(k >> 1) * 1024 + s * 2 + (k & 1)] = (k < NB_) ? spw[k * SEQ_ + s] : 0.0f;
}

// out[s,d] = sum_b fused[b,d] * spw[b,s] + spb[s]; K=512, all b64 loads, pipelined
__global__ void __launch_bounds__(256)
seq_gemm_kernel(const float* __restrict__ swz, const float* __restrict__ spb,
                const float* __restrict__ fswz, float* __restrict__ out) {
  int tid = threadIdx.x;
  int wave = tid >> 5, lane = tid & 31;
  int mt = blockIdx.x * 8 + wave;     // 32 m-tiles of s
  int m = lane & 15, g = lane >> 4;
  int s = mt * 16 + m;
  int n2 = m * 2;
  v8f acc[8] = {};
  v2f a_c = *(const v2f*)(swz + g * 1024 + s * 2);
  v2f b_c[8];
  #pragma unroll
  for (int nt = 0; nt < 8; ++nt)
    b_c[nt] = *(const v2f*)(fswz + g * 256 + nt * 32 + n2);
  for (int kb = 0; kb < 508; kb += 4) {
    int kp = (kb >> 1) + 2 + g;
    v2f a_n = *(const v2f*)(swz + (long)kp * 1024 + s * 2);
    v2f b_n[8];
    #pragma unroll
    for (int nt = 0; nt < 8; ++nt)
      b_n[nt] = *(const v2f*)(fswz + kp * 256 + nt * 32 + n2);
    #pragma unroll
    for (int nt = 0; nt < 8; ++nt) acc[nt] = wmma4(a_c, b_c[nt], acc[nt]);
    a_c = a_n;
    #pragma unroll
    for (int nt = 0; nt < 8; ++nt) b_c[nt] = b_n[nt];
  }
  #pragma unroll
  for (int nt = 0; nt < 8; ++nt) acc[nt] = wmma4(a_c, b_c[nt], acc[nt]);
  #pragma unroll
  for (int nt = 0; nt < 8; ++nt) {
    #pragma unroll
    for (int v = 0; v < 8; ++v) {
      int ss = mt * 16 + v + 8 * g;
      int d  = nt * 16 + m;
      out[ss * 128 + d] = acc[nt][v] + spb[ss];
    }
  }
}

extern "C" void kernel_launch(void* const* d_in, const int* in_sizes, int n_in,
                              void* d_out, int out_size, void* d_ws, size_t ws_size,
                              hipStream_t stream) {
  const int*   leidx   = (const int*)  d_in[0];
  const float* lwgt    = (const float*)d_in[1];
  const float* lkmer   = (const float*)d_in[2];
  const int*   lbatch  = (const int*)  d_in[3];
  const int*   geidx   = (const int*)  d_in[4];
  const float* gwgt    = (const float*)d_in[5];
  const float* gkmer   = (const float*)d_in[6];
  // 7,8,9: n_local, n_global, n_graphs (compile-time constants used instead)
  const float* lin_l_w = (const float*)d_in[10];
  const float* lin_l_b = (const float*)d_in[11];
  const float* ml_w1   = (const float*)d_in[12];
  const float* ml_b1   = (const float*)d_in[13];
  const float* ml_w2   = (const float*)d_in[14];
  const float* ml_b2   = (const float*)d_in[15];
  const float* lin_g_w = (const float*)d_in[16];
  const float* lin_g_b = (const float*)d_in[17];
  const float* mg_w1   = (const float*)d_in[18];
  const float* mg_b1   = (const float*)d_in[19];
  const float* mg_w2   = (const float*)d_in[20];
  const float* mg_b2   = (const float*)d_in[21];
  const float* attn_wv = (const float*)d_in[26];
  const float* attn_bv = (const float*)d_in[27];
  const float* attn_wo = (const float*)d_in[28];
  const float* attn_bo = (const float*)d_in[29];
  const float* proj_w  = (const float*)d_in[30];
  const float* proj_b  = (const float*)d_in[31];
  const float* spw     = (const float*)d_in[32];
  const float* spb     = (const float*)d_in[33];

  float* ws    = (float*)d_ws;
  float* agg_l = ws + WS_AGG_L;
  float* agg_g = ws + WS_AGG_G;
  float* pool  = ws + WS_POOL;
  float* cnt   = ws + WS_CNT;
  float* gsum  = ws + WS_GSUM;
  float* padd  = ws + WS_PADD;
  float* fswz  = ws + WS_FSWZ;
  float* spwz  = ws + WS_SPW;
  float* out   = (float*)d_out;

  zero_ws_kernel<<<4096, 256, 0, stream>>>(ws, WS_ZERO_N);
  spw_swz_kernel<<<(SEQ_ * SEQ_) / 256, 256, 0, stream>>>(spw, spwz);

  edge_msg_kernel<<<(EL + 255) / 256, 256, 0, stream>>>(leidx, lwgt, lkmer, lin_l_w, lin_l_b, agg_l, EL);
  edge_msg_kernel<<<(EG + 255) / 256, 256, 0, stream>>>(geidx, gwgt, gkmer, lin_g_w, lin_g_b, agg_g, EG);

  count_batch_kernel<<<1024, 256, 0, stream>>>(lbatch, cnt, (long)NL, NB_);

  node_mlp_kernel<0><<<(NL + 127) / 128, 256, 0, stream>>>(agg_l, ml_w1, ml_b1, ml_w2, ml_b2,
                                                           lbatch, pool, gsum, NL, NB_);
  node_mlp_kernel<1><<<NG / 128, 256, 0, stream>>>(agg_g, mg_w1, mg_b1, mg_w2, mg_b2,
                                                   lbatch, pool, gsum, NG, NB_);

  fuse_vec_kernel<<<1, 128, 0, stream>>>(gsum, attn_wv, attn_bv, attn_wo, attn_bo, proj_w, proj_b, padd);
  fuse_rows_kernel<<<(SEQ_ * DD) / 256, 256, 0, stream>>>(pool, cnt, padd, fswz, out + SEQ_ * DD);
  seq_gemm_kernel<<<4, 256, 0, stream>>>(spwz, spb, fswz, out);

  (void)in_sizes; (void)n_in; (void)out_size; (void)ws_size;
}